// SASILPConvLayer_75831942578725
// MI455X (gfx1250) — compile-verified
//
#include <hip/hip_runtime.h>
#include <hip/hip_bf16.h>

// CDNA5 / gfx1250, wave32. Problem constants from the reference:
//   N=100000 (multiple of 16), E=1600000, IN=OUT=128, R=8.

typedef float v2f __attribute__((ext_vector_type(2)));
typedef float v8f __attribute__((ext_vector_type(8)));

#define RGCN_R    8
#define RGCN_DIM  128
#define LDS_STR   132   // 128 + 4 pad: breaks 64-bank row aliasing for A reads

// ---------------------------------------------------------------------------
// Kernel 1: zero agg (N*128) and deg (N)
// ---------------------------------------------------------------------------
__global__ void rgcn_init(float* __restrict__ agg, float* __restrict__ deg, int N) {
    size_t i = (size_t)blockIdx.x * blockDim.x + threadIdx.x;
    size_t total = (size_t)N * RGCN_DIM;
    if (i < total) agg[i] = 0.0f;
    if (i < (size_t)N) deg[i] = 0.0f;
}

// ---------------------------------------------------------------------------
// Kernel 2: fused 9-way GEMM.
//   blockIdx.x -> node tile of 16 rows; blockIdx.y = rel (0..7 -> xw, 8 -> self)
//   256 threads = 8 waves; wave w computes the 16x16 tile for cols [16w,16w+16)
//   K = 128 -> 32 x v_wmma_f32_16x16x4_f32 per wave.
// ---------------------------------------------------------------------------
__global__ __launch_bounds__(256) void rgcn_gemm(
    const float* __restrict__ x,
    const float* __restrict__ rel_w,
    const float* __restrict__ w_self,
    const float* __restrict__ b_self,
    float* __restrict__ xw,
    float* __restrict__ out,
    int N)
{
    __shared__ float lds_x[16 * LDS_STR];

    const int tid   = threadIdx.x;
    const int node0 = blockIdx.x * 16;
    const int rel   = blockIdx.y;

    const float* __restrict__ W =
        (rel < RGCN_R) ? (rel_w + (size_t)rel * (RGCN_DIM * RGCN_DIM)) : w_self;

    // ---- stage 16x128 x-tile into LDS (512 float4 slots, 2 per thread) ----
#pragma unroll
    for (int j = 0; j < 2; ++j) {
        int s   = tid * 2 + j;       // 0..511
        int row = s >> 5;            // 32 float4 per row
        int c4  = s & 31;
        float4 v = ((const float4*)(x + (size_t)(node0 + row) * RGCN_DIM))[c4];
        float* p = &lds_x[row * LDS_STR + c4 * 4];
        p[0] = v.x; p[1] = v.y; p[2] = v.z; p[3] = v.w;
    }
    __syncthreads();

    const int lane = tid & 31;
    const int wave = tid >> 5;           // 0..7 -> 16-col group
    const int c0   = wave * 16;
    const int col  = c0 + (lane & 15);
    const int half = lane >> 4;          // upper half-wave handles K+2 / M+8
    const int koff = half * 2;
    const int arow = (lane & 15) * LDS_STR;

    // ---- preload all B fragments (W is K-major 128x128; hot in L2) ----
    // B layout (4x16, K x N): VGPR0 = K{koff}, VGPR1 = K{koff+1}, lane%16 = N.
    v2f b[32];
#pragma unroll
    for (int k = 0; k < 32; ++k) {
        const float* wp = W + (size_t)(4 * k + koff) * RGCN_DIM + col;
        b[k][0] = wp[0];
        b[k][1] = wp[RGCN_DIM];
    }

    // ---- K loop: A from LDS, accumulate with f32 WMMA ----
    v8f c = {0.f, 0.f, 0.f, 0.f, 0.f, 0.f, 0.f, 0.f};
#pragma unroll
    for (int k = 0; k < 32; ++k) {
        v2f a;
        a[0] = lds_x[arow + 4 * k + koff];
        a[1] = lds_x[arow + 4 * k + koff + 1];
        c = __builtin_amdgcn_wmma_f32_16x16x4_f32(
                /*neg_a=*/false, a, /*neg_b=*/false, b[k],
                /*c_mod=*/(short)0, c, /*reuse_a=*/false, /*reuse_b=*/false);
    }

    // ---- writeback: C VGPR v -> row v + 8*half, lane%16 -> col ----
    if (rel == RGCN_R) {
        const float bias = b_self[col];
#pragma unroll
        for (int v = 0; v < 8; ++v) {
            int row = v + half * 8;
            out[(size_t)(node0 + row) * RGCN_DIM + col] = c[v] + bias;
        }
    } else {
#pragma unroll
        for (int v = 0; v < 8; ++v) {
            int row = v + half * 8;
            xw[((size_t)(node0 + row) * RGCN_R + rel) * RGCN_DIM + col] = c[v];
        }
    }
}

// ---------------------------------------------------------------------------
// Kernel 3: per-edge gather + atomic scatter-add. One wave per edge:
//   lane loads float4 of xw[src, etype] and atomically adds into agg[dst].
// ---------------------------------------------------------------------------
__global__ __launch_bounds__(256) void rgcn_scatter(
    const int* __restrict__ edge_index,   // [2, E]: row0=src, row1=dst
    const int* __restrict__ edge_type,    // [E]
    const float* __restrict__ xw,
    float* __restrict__ agg,
    float* __restrict__ deg,
    int E)
{
    int gid  = blockIdx.x * blockDim.x + threadIdx.x;
    int e    = gid >> 5;
    int lane = gid & 31;
    if (e >= E) return;

    int src = edge_index[e];
    int dst = edge_index[E + e];
    int r   = edge_type[e];

    const float4* mp = (const float4*)(xw + ((size_t)src * RGCN_R + r) * RGCN_DIM);
    float4 v = mp[lane];

    float* ap = agg + (size_t)dst * RGCN_DIM + lane * 4;
    atomicAdd(ap + 0, v.x);
    atomicAdd(ap + 1, v.y);
    atomicAdd(ap + 2, v.z);
    atomicAdd(ap + 3, v.w);

    if (lane == 0) atomicAdd(deg + dst, 1.0f);
}

// ---------------------------------------------------------------------------
// Kernel 4: out = relu(out + agg / max(deg, 1))
// ---------------------------------------------------------------------------
__global__ void rgcn_finalize(float* __restrict__ out,
                              const float* __restrict__ agg,
                              const float* __restrict__ deg, int N)
{
    size_t i = (size_t)blockIdx.x * blockDim.x + threadIdx.x;
    if (i >= (size_t)N * RGCN_DIM) return;
    int n = (int)(i >> 7);
    float d = deg[n];
    d = d < 1.0f ? 1.0f : d;
    float v = out[i] + agg[i] / d;
    out[i] = v > 0.0f ? v : 0.0f;
}

// ---------------------------------------------------------------------------
extern "C" void kernel_launch(void* const* d_in, const int* in_sizes, int n_in,
                              void* d_out, int out_size, void* d_ws, size_t ws_size,
                              hipStream_t stream) {
    const float* x        = (const float*)d_in[0];
    const int*   eidx     = (const int*)d_in[1];
    const int*   etype    = (const int*)d_in[2];
    const float* rel_w    = (const float*)d_in[3];
    const float* w_self   = (const float*)d_in[4];
    const float* b_self   = (const float*)d_in[5];
    float*       out      = (float*)d_out;

    const int N = in_sizes[0] / RGCN_DIM;   // 100000 (multiple of 16)
    const int E = in_sizes[2];              // 1600000

    // workspace: xw [N,R,128] | agg [N,128] | deg [N]
    float* xw  = (float*)d_ws;
    float* agg = xw + (size_t)N * RGCN_R * RGCN_DIM;
    float* deg = agg + (size_t)N * RGCN_DIM;

    {   // zero agg + deg
        size_t total = (size_t)N * RGCN_DIM;
        int blocks = (int)((total + 255) / 256);
        rgcn_init<<<blocks, 256, 0, stream>>>(agg, deg, N);
    }
    {   // 9-way GEMM: rel 0..7 -> xw, rel 8 -> self term into out
        dim3 grid(N / 16, RGCN_R + 1);
        rgcn_gemm<<<grid, 256, 0, stream>>>(x, rel_w, w_self, b_self, xw, out, N);
    }
    {   // one wave per edge -> 8 edges per 256-thread block
        int blocks = (E + 7) / 8;
        rgcn_scatter<<<blocks, 256, 0, stream>>>(eidx, etype, xw, agg, deg, E);
    }
    {   // relu(out + agg/deg)
        size_t total = (size_t)N * RGCN_DIM;
        int blocks = (int)((total + 255) / 256);
        rgcn_finalize<<<blocks, 256, 0, stream>>>(out, agg, deg, N);
    }
}